// ResiduesPosEncode_51719996178662
// MI455X (gfx1250) — compile-verified
//
#include <hip/hip_runtime.h>
#include <hip/hip_bf16.h>

typedef float v2f __attribute__((ext_vector_type(2)));
typedef float v8f __attribute__((ext_vector_type(8)));

constexpr int NRES  = 102;
constexpr int TOTAL = 4284;   // 102 * (4+5+8+11+14)
constexpr int BATCH = 32;
constexpr int BG    = 5;      // batches per block: 5*3 = 15 of 16 WMMA columns used
constexpr int NBG   = (BATCH + BG - 1) / BG;   // 7

struct Params {
    const float* w[5];     // w_G..w_W  (16*natom, (natom+2)^2) row-major
    const int*   idx[5];   // (102, natom+2)
    const int*   pos[5];   // (102, natom)
};

__device__ __forceinline__ unsigned lds_off(const void* p) {
    // generic LDS address: high 32 bits = shared aperture, low 32 bits = LDS offset
    return (unsigned)(size_t)p;
}

__global__ __launch_bounds__(256)
void residues_encode_readout(const float* __restrict__ x,         // (32, 4284, 3)
                             const float* __restrict__ w_readout, // (128, 16)
                             Params P,
                             float* __restrict__ out)             // (32, 4284, 128, 3)
{
    __shared__ __align__(16) float s_xr[BG][16][3];   // gathered coords
    __shared__ __align__(16) float s_dxB[256][16];    // stage-1 B: dx[k][(bs,d)], zero k>=K
    __shared__ __align__(16) float s_enc[224][17];    // stage-1 D: enc[(a,c)][(bs,d)]
    __shared__ __align__(16) float s_outO[14][388];   // stage-2 D in output order [a][v*3+d]

    const int rg = blockIdx.x;          // 0..509 : letter-major residue id
    const int l  = rg / NRES;
    const int r  = rg % NRES;
    const int b0 = blockIdx.y * BG;
    const int nb = min(BG, BATCH - b0);

    int natom;
    switch (l) {
      case 0:  natom = 4;  break;  // G
      case 1:  natom = 5;  break;  // A
      case 2:  natom = 8;  break;  // L
      case 3:  natom = 11; break;  // F
      default: natom = 14; break;  // W
    }
    const int na2 = natom + 2;
    const int K   = na2 * na2;            // 36, 49, 100, 169, 256
    const int K4  = (K + 3) & ~3;         // zero-padded K rows in LDS
    const int KF  = K & ~3;               // branch-free main-loop bound
    const int rem = K - KF;               // 0 or 1 for these shapes

    const float* __restrict__ w   = P.w[l];
    const int*   __restrict__ idx = P.idx[l] + r * na2;
    const int*   __restrict__ pos = P.pos[l] + r * natom;

    const int tid   = threadIdx.x;
    const int wave  = tid >> 5;
    const int lane  = tid & 31;
    const int l15   = lane & 15;
    const int khalf = (lane >> 4) * 2;    // lanes 16..31 carry K+2, K+3 halves

    // ---- gather xr: per-lane async global->LDS scatter (CDNA5 async path) ----
    for (int i = tid; i < BG * 16 * 3; i += 256) {
        const int bs = i / 48, remx = i % 48, j = remx / 3, d = remx % 3;
        if (bs < nb && j < na2) {         // unloaded slots are never read
            const float* gp = &x[((size_t)(b0 + bs) * TOTAL + idx[j]) * 3 + d];
            const unsigned lo = lds_off(&s_xr[bs][j][d]);
            asm volatile("global_load_async_to_lds_b32 %0, %1, off"
                         :: "v"(lo), "v"(gp) : "memory");
        }
    }
    asm volatile("s_wait_asynccnt 0x0" ::: "memory");
    __syncthreads();

    // ---- build dx B-matrix: dx[k=(p,q)][(bs,d)] = xr[q][d] - xr[p][d] ----
    for (int i = tid; i < K4 * 16; i += 256) {
        const int k = i >> 4, col = i & 15;
        float v = 0.f;
        if (k < K && col < 15) {          // rows k>=K MUST be zero (K-tail); col 15 pad
            const int bs = col / 3, d = col % 3;
            const int p_ = k / na2, q_ = k % na2;
            v = s_xr[bs][q_][d] - s_xr[bs][p_][d];
        }
        s_dxB[k][col] = v;
    }

    // preload stage-2 A fragments (w_readout, invariant): 8 floats per lane
    const int vrow = wave * 16 + l15;
    v2f Awr[4];
    #pragma unroll
    for (int q = 0; q < 4; ++q) {
        Awr[q].x = w_readout[vrow * 16 + q * 4 + khalf];
        Awr[q].y = w_readout[vrow * 16 + q * 4 + khalf + 1];
    }
    __syncthreads();

    // ---- stage 1: enc[(a,c)][(bs,d)] = sum_k w[a*16+c][k] * dx[k][(bs,d)]
    //      one 16-row M-tile per atom; branch-free K loop + uniform tail
    for (int a = wave; a < natom; a += 8) {        // wave-uniform: EXEC stays full
        v8f acc = {};
        const int row = a * 16 + l15;
        const float* __restrict__ wp = w + (size_t)row * K + khalf;
        for (int k0 = 0; k0 < KF; k0 += 4) {
            v2f A, B;
            A.x = wp[k0];
            A.y = wp[k0 + 1];
            B.x = s_dxB[k0 + khalf][l15];
            B.y = s_dxB[k0 + khalf + 1][l15];
            acc = __builtin_amdgcn_wmma_f32_16x16x4_f32(false, A, false, B,
                                                        (short)0, acc, false, false);
        }
        if (rem) {  // uniform tail: A addresses clamped, B rows are zero-padded
            const int Km1 = K - 1;
            v2f A, B;
            A.x = w[(size_t)row * K + min(KF + khalf,     Km1)];
            A.y = w[(size_t)row * K + min(KF + khalf + 1, Km1)];
            B.x = s_dxB[KF + khalf][l15];
            B.y = s_dxB[KF + khalf + 1][l15];
            acc = __builtin_amdgcn_wmma_f32_16x16x4_f32(false, A, false, B,
                                                        (short)0, acc, false, false);
        }
        const int mb = (lane >> 4) * 8;            // D layout: VGPR p -> M=p / M=p+8
        #pragma unroll
        for (int p_ = 0; p_ < 8; ++p_)
            s_enc[a * 16 + mb + p_][l15] = acc[p_];
    }

    const int ntiles = (natom * 3 + 15) / 16;      // 1, 1, 2, 3, 3

    for (int bs = 0; bs < nb; ++bs) {
        asm volatile("s_wait_asynccnt 0x0" ::: "memory"); // s_outO drained
        __syncthreads();                                   // enc ready / outO reusable

        // ---- stage-2 WMMA: out[v][(a,d)] = sum_c w_readout[v][c] * enc[(a,c)][(bs,d)]
        const int bcol = bs * 3;
        for (int nt = 0; nt < ntiles; ++nt) {
            const int n    = nt * 16 + l15;
            const int a16  = n / 3;
            const int dd   = n % 3;
            const int amin = min(a16, natom - 1); // clamp pad lanes (discarded below)
            const int erow = amin * 16;
            v8f acc = {};
            #pragma unroll
            for (int q = 0; q < 4; ++q) {
                const int ka = q * 4 + khalf;
                v2f B;
                B.x = s_enc[erow + ka][bcol + dd];
                B.y = s_enc[erow + ka + 1][bcol + dd];
                acc = __builtin_amdgcn_wmma_f32_16x16x4_f32(false, Awr[q], false, B,
                                                            (short)0, acc, false, false);
            }
            // write D directly in output order: s_outO[a][v*3+d]
            if (a16 < natom) {                     // pad lanes: no LDS write
                const int mb = (lane >> 4) * 8;
                #pragma unroll
                for (int p_ = 0; p_ < 8; ++p_)
                    s_outO[a16][(wave * 16 + mb + p_) * 3 + dd] = acc[p_];
            }
        }
        __syncthreads();

        // ---- output: per-lane async LDS->global 128-bit DMA, fully coalesced ----
        const int b   = b0 + bs;
        const int nf4 = natom * 96;                // float4s per batch (384 f per atom)
        for (int i = tid; i < nf4; i += 256) {
            const int a = i / 96, f4 = i % 96;
            const unsigned lo = lds_off(&s_outO[a][f4 * 4]);
            const float* gp = out + ((size_t)b * TOTAL + pos[a]) * 384 + f4 * 4;
            asm volatile("global_store_async_from_lds_b128 %0, %1, off"
                         :: "v"(gp), "v"(lo) : "memory");
        }
        // next iteration (or implicit wait-idle at S_ENDPGM) drains ASYNCcnt
    }
}

extern "C" void kernel_launch(void* const* d_in, const int* in_sizes, int n_in,
                              void* d_out, int out_size, void* d_ws, size_t ws_size,
                              hipStream_t stream) {
    // setup_inputs() order: x, w_G, w_A, w_L, w_F, w_W, w_readout,
    //                       idx_G, pos_G, idx_A, pos_A, idx_L, pos_L,
    //                       idx_F, pos_F, idx_W, pos_W
    const float* x  = (const float*)d_in[0];
    Params P;
    P.w[0] = (const float*)d_in[1];
    P.w[1] = (const float*)d_in[2];
    P.w[2] = (const float*)d_in[3];
    P.w[3] = (const float*)d_in[4];
    P.w[4] = (const float*)d_in[5];
    const float* wr = (const float*)d_in[6];
    P.idx[0] = (const int*)d_in[7];  P.pos[0] = (const int*)d_in[8];
    P.idx[1] = (const int*)d_in[9];  P.pos[1] = (const int*)d_in[10];
    P.idx[2] = (const int*)d_in[11]; P.pos[2] = (const int*)d_in[12];
    P.idx[3] = (const int*)d_in[13]; P.pos[3] = (const int*)d_in[14];
    P.idx[4] = (const int*)d_in[15]; P.pos[4] = (const int*)d_in[16];

    dim3 grid(5 * NRES, NBG);   // (510 residues) x (7 batch groups of 5)
    residues_encode_readout<<<grid, 256, 0, stream>>>(x, wr, P, (float*)d_out);
}